// DecorrelatedBN_20392504721409
// MI455X (gfx1250) — compile-verified
//
#include <hip/hip_runtime.h>
#include <hip/hip_bf16.h>

typedef __attribute__((ext_vector_type(2))) float v2f;
typedef __attribute__((ext_vector_type(8))) float v8f;

#define M_TOTAL   65536      // 64*32*32 rows
#define C_TOTAL   512
#define GROUPS    8
#define GSIZE     64
#define INV_M     (1.0f / 65536.0f)
#define EPS       1e-5f
#define NS_ITERS  12
#define COV_ROWS  2048       // rows per cov block (2 waves of 1024 per i-tile)

// ---------------------------------------------------------------------------
// WMMA helper: D = A(16x4) * B(4x16) + C, fp32
// ---------------------------------------------------------------------------
__device__ __forceinline__ v8f wmma_f32(v2f a, v2f b, v8f c) {
    return __builtin_amdgcn_wmma_f32_16x16x4_f32(
        /*neg_a=*/false, a, /*neg_b=*/false, b,
        /*c_mod=*/(short)0, c, /*reuse_a=*/false, /*reuse_b=*/false);
}

// ---------------------------------------------------------------------------
// Kernel 1: per-group Gram matrix S[g] = sum_m x_m x_m^T  (WMMA)
// Channel sums computed via an extra ones-A WMMA (D = 1^T * B -> column sums),
// so x is never re-read for the mean.
// grid = GROUPS * (M_TOTAL/COV_ROWS) blocks of 256 threads (8 waves).
// Wave w: i-tile = w&3 (16 channels), row-half = w>>2 (COV_ROWS/2 rows).
// Row-half pairs are reduced through LDS before global atomics.
// ---------------------------------------------------------------------------
__global__ void __launch_bounds__(256)
cov_kernel(const float* __restrict__ x,
           float* __restrict__ S,          // [G][64][64]
           float* __restrict__ sums)       // [512]
{
    const int g     = blockIdx.x & 7;
    const int chunk = blockIdx.x >> 3;
    const int r0    = chunk * COV_ROWS;
    const int gbase = g * GSIZE;
    const int tid   = threadIdx.x;

    const int wave  = tid >> 5;
    const int lane  = tid & 31;
    const int itile = wave & 3;              // 16-channel row-slab of S
    const int rhalf = wave >> 2;             // 0/1 -> which row half
    const int li    = lane & 15;
    const int kh    = lane >> 4;             // 0/1

    v8f acc[4] = {};                         // tile t -> j-tile (itile+t)&3
    v8f accS = {};                           // ones^T * A -> channel sums
    const v2f ones = {1.0f, 1.0f};

    const int rowStart = r0 + rhalf * (COV_ROWS / 2);
    const float* xg = x + gbase;

    for (int k0 = 0; k0 < COV_ROWS / 2; k0 += 4) {
        const float* pa = xg + (size_t)(rowStart + k0 + 2 * kh) * C_TOTAL;
        // A fragment: A[i=li][k] = x[row_k][gbase + itile*16 + li]
        v2f A;
        A.x = pa[itile * 16 + li];
        A.y = pa[C_TOTAL + itile * 16 + li];
        acc[0] = wmma_f32(A, A, acc[0]);             // diagonal tile: B == A
        accS   = wmma_f32(ones, A, accS);            // channel sums (free)
#pragma unroll
        for (int t = 1; t < 4; ++t) {
            const int jt = (itile + t) & 3;
            v2f B;
            B.x = pa[jt * 16 + li];
            B.y = pa[C_TOTAL + jt * 16 + li];
            acc[t] = wmma_f32(A, B, acc[t]);
        }
    }

    // channel sums: every row of accS holds the same column sums; take M=0.
    if (kh == 0)
        atomicAdd(&sums[gbase + itile * 16 + li], accS[0]);

    // pair-reduce the two row-half waves through LDS, then atomics from waves 0-3
    __shared__ float redbuf[4][4][32][8];            // 16 KB
    if (wave >= 4) {
#pragma unroll
        for (int t = 0; t < 4; ++t)
            *(v8f*)&redbuf[wave - 4][t][lane][0] = acc[t];
    }
    __syncthreads();
    if (wave < 4) {
        float* Sg = S + g * (GSIZE * GSIZE);
#pragma unroll
        for (int t = 0; t < 4; ++t) {
            acc[t] += *(const v8f*)&redbuf[wave][t][lane][0];
            const int jt = (itile + t) & 3;
#pragma unroll
            for (int r = 0; r < 8; ++r) {
                const int mrow = r + 8 * kh;         // D layout: lanes16-31 -> M=r+8
                atomicAdd(&Sg[(itile * 16 + mrow) * GSIZE + jt * 16 + li], acc[t][r]);
            }
        }
    }
}

// ---------------------------------------------------------------------------
// Kernel 2: per-group solver.  sigma = S/M - mu mu^T + eps*I,
// P = sigma^{-1/2} via coupled Newton-Schulz with Frobenius normalization.
// grid = 8 blocks x 256 threads; 48KB LDS.
// ---------------------------------------------------------------------------
__device__ __forceinline__ void mat_3I_minus_LR(float* __restrict__ T,
                                                const float* __restrict__ L,
                                                const float* __restrict__ R,
                                                int tid)
{
    const int i0 = (tid >> 4) << 2;
    const int j0 = (tid & 15) << 2;
    float a[4][4] = {};
    for (int k = 0; k < 64; ++k) {
        const float l0 = L[(i0 + 0) * 64 + k];
        const float l1 = L[(i0 + 1) * 64 + k];
        const float l2 = L[(i0 + 2) * 64 + k];
        const float l3 = L[(i0 + 3) * 64 + k];
        const float4 r = *(const float4*)&R[k * 64 + j0];
        a[0][0] += l0 * r.x; a[0][1] += l0 * r.y; a[0][2] += l0 * r.z; a[0][3] += l0 * r.w;
        a[1][0] += l1 * r.x; a[1][1] += l1 * r.y; a[1][2] += l1 * r.z; a[1][3] += l1 * r.w;
        a[2][0] += l2 * r.x; a[2][1] += l2 * r.y; a[2][2] += l2 * r.z; a[2][3] += l2 * r.w;
        a[3][0] += l3 * r.x; a[3][1] += l3 * r.y; a[3][2] += l3 * r.z; a[3][3] += l3 * r.w;
    }
#pragma unroll
    for (int ii = 0; ii < 4; ++ii)
#pragma unroll
        for (int jj = 0; jj < 4; ++jj)
            T[(i0 + ii) * 64 + j0 + jj] =
                ((i0 + ii) == (j0 + jj) ? 3.0f : 0.0f) - a[ii][jj];
}

__device__ __forceinline__ void ns_update(float* __restrict__ Y,
                                          float* __restrict__ Z,
                                          const float* __restrict__ T,
                                          int tid)
{
    const int i0 = (tid >> 4) << 2;
    const int j0 = (tid & 15) << 2;
    float ay[4][4] = {};
    float az[4][4] = {};
    for (int k = 0; k < 64; ++k) {
        const float y0 = Y[(i0 + 0) * 64 + k];
        const float y1 = Y[(i0 + 1) * 64 + k];
        const float y2 = Y[(i0 + 2) * 64 + k];
        const float y3 = Y[(i0 + 3) * 64 + k];
        const float t0 = T[(i0 + 0) * 64 + k];
        const float t1 = T[(i0 + 1) * 64 + k];
        const float t2 = T[(i0 + 2) * 64 + k];
        const float t3 = T[(i0 + 3) * 64 + k];
        const float4 tr = *(const float4*)&T[k * 64 + j0];
        const float4 zr = *(const float4*)&Z[k * 64 + j0];
        ay[0][0] += y0 * tr.x; ay[0][1] += y0 * tr.y; ay[0][2] += y0 * tr.z; ay[0][3] += y0 * tr.w;
        ay[1][0] += y1 * tr.x; ay[1][1] += y1 * tr.y; ay[1][2] += y1 * tr.z; ay[1][3] += y1 * tr.w;
        ay[2][0] += y2 * tr.x; ay[2][1] += y2 * tr.y; ay[2][2] += y2 * tr.z; ay[2][3] += y2 * tr.w;
        ay[3][0] += y3 * tr.x; ay[3][1] += y3 * tr.y; ay[3][2] += y3 * tr.z; ay[3][3] += y3 * tr.w;
        az[0][0] += t0 * zr.x; az[0][1] += t0 * zr.y; az[0][2] += t0 * zr.z; az[0][3] += t0 * zr.w;
        az[1][0] += t1 * zr.x; az[1][1] += t1 * zr.y; az[1][2] += t1 * zr.z; az[1][3] += t1 * zr.w;
        az[2][0] += t2 * zr.x; az[2][1] += t2 * zr.y; az[2][2] += t2 * zr.z; az[2][3] += t2 * zr.w;
        az[3][0] += t3 * zr.x; az[3][1] += t3 * zr.y; az[3][2] += t3 * zr.z; az[3][3] += t3 * zr.w;
    }
    __syncthreads();   // all reads of Y/Z done before overwrite
#pragma unroll
    for (int ii = 0; ii < 4; ++ii)
#pragma unroll
        for (int jj = 0; jj < 4; ++jj) {
            Y[(i0 + ii) * 64 + j0 + jj] = 0.5f * ay[ii][jj];
            Z[(i0 + ii) * 64 + j0 + jj] = 0.5f * az[ii][jj];
        }
}

__global__ void __launch_bounds__(256)
solver_kernel(const float* __restrict__ S,
              const float* __restrict__ sums,
              float* __restrict__ P,        // [G][64][64]
              float* __restrict__ meanOut)  // [512]
{
    const int g   = blockIdx.x;
    const int tid = threadIdx.x;

    __shared__ float Yb[64 * 64];
    __shared__ float Zb[64 * 64];
    __shared__ float Tb[64 * 64];
    __shared__ float meanl[64];
    __shared__ float red[256];

    if (tid < 64) {
        const float mu = sums[g * 64 + tid] * INV_M;
        meanl[tid] = mu;
        meanOut[g * 64 + tid] = mu;
    }
    __syncthreads();

    // sigma + eps*I into Tb, accumulate Frobenius^2
    float fr = 0.0f;
    for (int e = tid; e < 4096; e += 256) {
        const int i = e >> 6, j = e & 63;
        float v = S[g * 4096 + e] * INV_M - meanl[i] * meanl[j];
        if (i == j) v += EPS;
        Tb[e] = v;
        fr += v * v;
    }
    red[tid] = fr;
    __syncthreads();
    for (int s = 128; s > 0; s >>= 1) {
        if (tid < s) red[tid] += red[tid + s];
        __syncthreads();
    }
    const float fnorm = sqrtf(red[0]);
    const float inv_f = 1.0f / fnorm;

    // Y0 = A/f, Z0 = I
    for (int e = tid; e < 4096; e += 256) {
        Yb[e] = Tb[e] * inv_f;
        Zb[e] = ((e >> 6) == (e & 63)) ? 1.0f : 0.0f;
    }
    __syncthreads();

    for (int it = 0; it < NS_ITERS; ++it) {
        mat_3I_minus_LR(Tb, Zb, Yb, tid);   // T = 3I - Z*Y
        __syncthreads();
        ns_update(Yb, Zb, Tb, tid);         // Y=0.5*Y*T ; Z=0.5*T*Z
        __syncthreads();
    }

    // A^{-1/2} = Z / sqrt(f)
    const float sc = rsqrtf(fnorm);
    for (int e = tid; e < 4096; e += 256)
        P[g * 4096 + e] = Zb[e] * sc;
}

// ---------------------------------------------------------------------------
// Kernel 3: out = ((x - mu) @ P) * gamma + beta   (WMMA, P staged in LDS)
// grid = (M_TOTAL/128) * GROUPS blocks x 256 threads.
// Block: group g = blockIdx&7, rows [rb*128, rb*128+128). Wave w -> 16 rows.
// ---------------------------------------------------------------------------
__global__ void __launch_bounds__(256)
apply_kernel(const float* __restrict__ x,
             const float* __restrict__ P,
             const float* __restrict__ mean,
             const float* __restrict__ gamma,
             const float* __restrict__ beta,
             float* __restrict__ out)
{
    const int g       = blockIdx.x & 7;
    const int rowBase = (blockIdx.x >> 3) * 128;
    const int gbase   = g * GSIZE;
    const int tid     = threadIdx.x;

    __shared__ float Pl[64 * 64];
    __shared__ float meanl[64];

    {
        const float4* src = (const float4*)(P + g * 4096);
        float4* dst = (float4*)Pl;
        for (int i = tid; i < 1024; i += 256) dst[i] = src[i];
        if (tid < 64) meanl[tid] = mean[gbase + tid];
    }
    __syncthreads();

    const int wave = tid >> 5;
    const int lane = tid & 31;
    const int li   = lane & 15;
    const int kh   = lane >> 4;              // 0/1
    const int row0 = rowBase + wave * 16;

    v8f acc[4] = {};
    const float* xrow = x + (size_t)(row0 + li) * C_TOTAL + gbase;

    for (int i0 = 0; i0 < 64; i0 += 4) {
        const int ch = i0 + 2 * kh;
        // A fragment: A[m=li][k] = centered x[row0+li][ch..ch+1]  (one b64 load)
        const float2 av = *(const float2*)(xrow + ch);
        v2f A;
        A.x = av.x - meanl[ch];
        A.y = av.y - meanl[ch + 1];
#pragma unroll
        for (int jt = 0; jt < 4; ++jt) {
            // B fragment: B[k][n=li] = P[ch..ch+1][jt*16+li]
            v2f B;
            B.x = Pl[ch * 64 + jt * 16 + li];
            B.y = Pl[(ch + 1) * 64 + jt * 16 + li];
            acc[jt] = wmma_f32(A, B, acc[jt]);
        }
    }

#pragma unroll
    for (int jt = 0; jt < 4; ++jt) {
        const int col = gbase + jt * 16 + li;
        const float gm = gamma[col];
        const float bt = beta[col];
#pragma unroll
        for (int r = 0; r < 8; ++r) {
            const int row = row0 + r + 8 * kh;   // D layout: lanes16-31 -> M=r+8
            out[(size_t)row * C_TOTAL + col] = acc[jt][r] * gm + bt;
        }
    }
}

// ---------------------------------------------------------------------------
// Host-side launch
// ---------------------------------------------------------------------------
extern "C" void kernel_launch(void* const* d_in, const int* in_sizes, int n_in,
                              void* d_out, int out_size, void* d_ws, size_t ws_size,
                              hipStream_t stream)
{
    const float* x     = (const float*)d_in[0];
    const float* gamma = (const float*)d_in[1];
    const float* beta  = (const float*)d_in[2];
    float* out = (float*)d_out;

    // Workspace layout (floats):
    //   [0, 32768)        S     (G x 64 x 64 Gram accumulators)
    //   [32768, 33280)    sums  (512 channel sums)
    //   [33280, 66048)    P     (G x 64 x 64 projections)
    //   [66048, 66560)    mean  (512)
    float* ws    = (float*)d_ws;
    float* ws_S  = ws;
    float* ws_su = ws + 32768;
    float* ws_P  = ws + 33280;
    float* ws_mu = ws + 66048;

    // Zero the accumulators (S + sums) every launch.
    hipMemsetAsync(d_ws, 0, (size_t)(32768 + 512) * sizeof(float), stream);

    cov_kernel<<<GROUPS * (M_TOTAL / COV_ROWS), 256, 0, stream>>>(x, ws_S, ws_su);
    solver_kernel<<<GROUPS, 256, 0, stream>>>(ws_S, ws_su, ws_P, ws_mu);
    apply_kernel<<<(M_TOTAL / 128) * GROUPS, 256, 0, stream>>>(x, ws_P, ws_mu,
                                                               gamma, beta, out);
}